// FirstOrderCondRNN_84172769067906
// MI455X (gfx1250) — compile-verified
//
#include <hip/hip_runtime.h>

// ---- fixed problem dims from the reference setup ----
#define B_    48
#define NKC   800
#define NM    20      // n_mbon == n_dan
#define NF    60      // n_fbn
#define ND    20      // n_dan
#define NR    100     // n_recur
#define T_    121
#define PLANE (NM * NKC)   // 16000
#define W_MAXV 0.05f

typedef float v2f __attribute__((ext_vector_type(2)));
typedef float v4f __attribute__((ext_vector_type(4)));
typedef float v8f __attribute__((ext_vector_type(8)));

__launch_bounds__(1024, 1)
__global__ void cond_rnn_kernel(const float* __restrict__ r_kc,     // (B, NKC, T)
                                const float* __restrict__ r_ext,    // (B, 2, T)
                                const float* __restrict__ time_v,   // (T)
                                const float* __restrict__ W_kc0,    // (B, NM, NKC)
                                const float* __restrict__ wt0,      // (B, NM, NKC)
                                const float* __restrict__ W_recur,  // (NR, NR)
                                const float* __restrict__ W_readout,// (1, NM)
                                const float* __restrict__ bias,     // (NR)
                                const float* __restrict__ W_ext,    // (NF, 2)
                                float* __restrict__ r_all,          // (T, B, NR)
                                float* __restrict__ W_all,          // (T, B, NM, NKC)
                                float* __restrict__ wt_all,         // (T, B, NM, NKC)
                                float* __restrict__ ro_all)         // (T, B)
{
    const int b    = blockIdx.x;
    const int tid  = threadIdx.x;
    const int lane = tid & 31;          // wave32
    const int wv   = tid >> 5;

    // ---- LDS-resident per-batch state (~141 KB) ----
    __shared__ __align__(16) float sW[PLANE];         // 64000 B
    __shared__ __align__(16) float sWt[PLANE];        // 64000 B
    __shared__ __align__(16) float sRkc[2][NKC];      // double-buffered input column
    __shared__ __align__(16) float sRbKc[NKC];
    __shared__ float sR[NR];
    __shared__ float sRbDan[ND];
    __shared__ float sImPart[8][NM];    // per-wave WMMA partials
    __shared__ float sIfbn[NF];
    __shared__ float sWrp[NR];
    __shared__ float sBias[NR];
    __shared__ float sWext[NF * 2];
    __shared__ float sWro[NM];
    __shared__ float sRext[2][2];

    const float dtv = time_v[1] - time_v[0];   // 0.5
    const float aR  = dtv;                     // dt / TAU_R (TAU_R = 1)
    const float aW  = dtv * 0.2f;              // dt / TAU_W (TAU_W = 5)
    const size_t rkcBase = (size_t)b * NKC * T_;

    // ---------------- init + t=0 outputs ----------------
    if (tid < 1000) {       // 16 contiguous elems/thread, no row crossing (16 | 800)
        const int e0 = tid * 16;
        #pragma unroll
        for (int c = 0; c < 4; ++c) {
            const int idx = e0 + c * 4;
            v4f w0 = *(const v4f*)&W_kc0[(size_t)b * PLANE + idx];
            v4f t0 = *(const v4f*)&wt0  [(size_t)b * PLANE + idx];
            *(v4f*)&sW[idx]  = w0;
            *(v4f*)&sWt[idx] = t0;
            __builtin_nontemporal_store(w0, (v4f*)&W_all [(size_t)b * PLANE + idx]);
            __builtin_nontemporal_store(t0, (v4f*)&wt_all[(size_t)b * PLANE + idx]);
        }
    }
    if (tid < NR) {
        float r0 = (tid < NM) ? 0.0f : 0.1f;
        sR[tid]    = r0;
        sBias[tid] = bias[tid];
        __builtin_nontemporal_store(r0, &r_all[(size_t)b * NR + tid]);
    }
    if (tid < NKC) {        // rkc at t=0 is also rb_kc0
        float v = r_kc[rkcBase + (size_t)tid * T_];
        sRkc[0][tid] = v;
        sRbKc[tid]   = v;
    }
    if (tid < ND)     sRbDan[tid] = 0.1f;
    if (tid < NF * 2) sWext[tid]  = W_ext[tid];
    if (tid < NM)     sWro[tid]   = W_readout[tid];
    if (tid < 2)      sRext[0][tid] = r_ext[(size_t)b * 2 * T_ + (size_t)tid * T_];
    if (tid == 0)     __builtin_nontemporal_store(0.0f, &ro_all[b]); // mbon part of r0 is 0
    __syncthreads();

    // ---------------- 120 sequential steps ----------------
    for (int t = 0; t < T_ - 1; ++t) {
        const int cur = t & 1;
        const int nxt = cur ^ 1;

        // P1a (waves 0-7): I_mbon = W(20x800) . rkc(800) via V_WMMA_F32_16X16X4_F32.
        // B fragment replicated over N; K split across 8 waves (25 k-steps each).
        if (wv < 8) {
            v8f c0 = {0.f,0.f,0.f,0.f,0.f,0.f,0.f,0.f};
            v8f c1 = {0.f,0.f,0.f,0.f,0.f,0.f,0.f,0.f};
            const int row0  = lane & 15;                 // M tile 0: rows 0..15
            const int row1  = row0 + 16;                 // M tile 1: rows 16..31
            const int arow1 = (row1 < NM) ? row1 : row0; // clamp addr; garbage D rows unused
            const int koff  = (lane >> 4) << 1;          // K = koff + vgpr_j
            const float* pA0 = &sW[row0  * NKC + koff];
            const float* pA1 = &sW[arow1 * NKC + koff];
            const float* pB  = &sRkc[cur][koff];
            const int kbase = wv * 100;
            for (int kk = 0; kk < 100; kk += 4) {
                const int k = kbase + kk;
                v2f a0 = *(const v2f*)(pA0 + k);
                v2f a1 = *(const v2f*)(pA1 + k);
                v2f bb = *(const v2f*)(pB  + k);
                c0 = __builtin_amdgcn_wmma_f32_16x16x4_f32(false, a0, false, bb,
                                                           (short)0, c0, false, false);
                c1 = __builtin_amdgcn_wmma_f32_16x16x4_f32(false, a1, false, bb,
                                                           (short)0, c1, false, false);
            }
            // column N=0: lane 0 holds M=0..7 (tile0) / M=16..19 (tile1); lane 16 holds M=8..15
            if (lane == 0) {
                #pragma unroll
                for (int i = 0; i < 8; ++i) sImPart[wv][i] = c0[i];
                #pragma unroll
                for (int i = 0; i < 4; ++i) sImPart[wv][16 + i] = c1[i];
            } else if (lane == 16) {
                #pragma unroll
                for (int i = 0; i < 8; ++i) sImPart[wv][8 + i] = c0[i];
            }
        }
        // P1b (waves 8-11): Wr_prod[j] = sum_i Wr[j,i] * r[i], with Wr[:NM, NR-ND:] == 0
        else if (wv < 12) {
            const int j = tid - 256;
            if (j < NR) {
                const v4f* wrow = (const v4f*)(W_recur + (size_t)j * NR); // 400B rows, 16B aligned
                const int lim4 = (j < NM) ? ((NR - ND) / 4) : (NR / 4);
                float s = 0.0f;
                for (int i4 = 0; i4 < lim4; ++i4) {
                    v4f w = wrow[i4];
                    const int i = i4 * 4;
                    s += w[0] * sR[i] + w[1] * sR[i + 1] + w[2] * sR[i + 2] + w[3] * sR[i + 3];
                }
                sWrp[j] = s;
            }
        }
        // P1c (waves 12-13): I_fbn = W_ext(60x2) . rext(2)
        else if (wv < 14) {
            const int f = tid - 384;
            if (f < NF) sIfbn[f] = sRext[cur][0] * sWext[f * 2] + sRext[cur][1] * sWext[f * 2 + 1];
        }
        // P1d (waves 14-31): prefetch next step's strided r_kc column into other buffer
        else {
            if (t + 1 < T_ - 1) {
                const int k = tid - 448;     // 0..575
                sRkc[nxt][k] = r_kc[rkcBase + (size_t)k * T_ + (t + 1)];
                if (k < NKC - 576)           // 0..223 -> covers 576..799
                    sRkc[nxt][k + 576] = r_kc[rkcBase + (size_t)(k + 576) * T_ + (t + 1)];
                if (k < 2)
                    sRext[nxt][k] = r_ext[(size_t)b * 2 * T_ + (size_t)k * T_ + (t + 1)];
            }
        }
        __syncthreads();

        // P2: recurrent state update r_new = r + (relu(Wr_prod + bias + I) - r) * dt
        if (tid < NR) {
            float Itot;
            if (tid < NM) {
                float s = 0.0f;
                #pragma unroll
                for (int w = 0; w < 8; ++w) s += sImPart[w][tid];
                Itot = s;
            } else if (tid < NM + NF) {
                Itot = sIfbn[tid - NM];
            } else {
                Itot = 0.0f;
            }
            float pre = sWrp[tid] + sBias[tid] + Itot;
            float act = pre > 0.0f ? pre : 0.0f;
            float rn  = sR[tid] + (act - sR[tid]) * aR;
            sR[tid] = rn;
            __builtin_nontemporal_store(rn, &r_all[((size_t)(t + 1) * B_ + b) * NR + tid]);
        }
        __syncthreads();

        // P3: readout + slow traces
        if (tid == 0) {
            float s = 0.0f;
            #pragma unroll
            for (int m = 0; m < NM; ++m) s += sR[m] * sWro[m];
            __builtin_nontemporal_store(s, &ro_all[(size_t)(t + 1) * B_ + b]);
        }
        if (tid < NKC) sRbKc[tid] += (sRkc[cur][tid] - sRbKc[tid]) * aW;
        if (tid >= NKC && tid < NKC + ND) {
            const int d = tid - NKC;
            sRbDan[d] += (sR[NR - ND + d] - sRbDan[d]) * aW;
        }
        __syncthreads();

        // P4: plasticity plane update + streamed history writes (bandwidth-dominant)
        if (tid < 1000) {
            const int d  = tid / 50;                // row, fixed per thread
            const int k0 = (tid - d * 50) * 16;     // 16 contiguous cols
            const float rbd = sRbDan[d];
            const float rdn = sR[NR - ND + d];
            const size_t gb = ((size_t)(t + 1) * B_ + b) * (size_t)PLANE + (size_t)d * NKC + k0;
            const int l0 = d * NKC + k0;
            #pragma unroll
            for (int c = 0; c < 4; ++c) {
                const int k = k0 + c * 4;
                const int l = l0 + c * 4;
                v4f rk  = *(const v4f*)&sRkc[cur][k];
                v4f rb  = *(const v4f*)&sRbKc[k];
                v4f wt4 = *(const v4f*)&sWt[l];
                v4f w4  = *(const v4f*)&sW[l];
                v4f wtn, wn;
                #pragma unroll
                for (int q = 0; q < 4; ++q) {
                    float dw = rbd * rk[q] - rdn * rb[q];
                    float wtv = wt4[q] + dw * dtv;
                    float x = w4[q] + (wtv - w4[q]) * aW;
                    x = fminf(fmaxf(x, 0.0f), W_MAXV);
                    wtn[q] = wtv;
                    wn[q]  = x;
                }
                *(v4f*)&sWt[l] = wtn;
                *(v4f*)&sW[l]  = wn;
                __builtin_nontemporal_store(wn,  (v4f*)&W_all [gb + c * 4]);
                __builtin_nontemporal_store(wtn, (v4f*)&wt_all[gb + c * 4]);
            }
        }
        __syncthreads();   // protects sW/sWt/sRkc for next iteration
    }
}

extern "C" void kernel_launch(void* const* d_in, const int* in_sizes, int n_in,
                              void* d_out, int out_size, void* d_ws, size_t ws_size,
                              hipStream_t stream) {
    (void)in_sizes; (void)n_in; (void)out_size; (void)d_ws; (void)ws_size;
    const float* r_kc      = (const float*)d_in[0];
    const float* r_ext     = (const float*)d_in[1];
    const float* time_v    = (const float*)d_in[2];
    const float* W_kc0     = (const float*)d_in[3];
    const float* wt0       = (const float*)d_in[4];
    const float* W_recur   = (const float*)d_in[5];
    const float* W_readout = (const float*)d_in[6];
    const float* bias      = (const float*)d_in[7];
    const float* W_ext     = (const float*)d_in[8];
    // d_in[9] = n_batch (known: 48)

    float* out    = (float*)d_out;
    float* r_all  = out;                                    // T*B*NR
    float* W_all  = r_all  + (size_t)T_ * B_ * NR;          // T*B*NM*NKC
    float* wt_all = W_all  + (size_t)T_ * B_ * PLANE;       // T*B*NM*NKC
    float* ro_all = wt_all + (size_t)T_ * B_ * PLANE;       // T*B

    hipLaunchKernelGGL(cond_rnn_kernel, dim3(B_), dim3(1024), 0, stream,
                       r_kc, r_ext, time_v, W_kc0, wt0, W_recur, W_readout,
                       bias, W_ext, r_all, W_all, wt_all, ro_all);
}